// GraphConvolutionBlock_33440615367246
// MI455X (gfx1250) — compile-verified
//
#include <hip/hip_runtime.h>
#include <hip/hip_bf16.h>

typedef __attribute__((ext_vector_type(2))) float v2f;
typedef __attribute__((ext_vector_type(8))) float v8f;

#define NB 8
#define CIN 64
#define LL 4096
#define NN (NB * LL)        // 32768
#define COUT 64
#define HEADS 3
#define HC (HEADS * COUT)   // 192
#define EE 524288
#define NEG 0.2f

// ---------------- float atomic max (signed-int / unsigned-min trick) ----------
__device__ inline void atomicMaxFloat(float* addr, float value) {
    if (value >= 0.0f) {
        atomicMax((int*)addr, __float_as_int(value));
    } else {
        atomicMin((unsigned int*)addr, __float_as_uint(value));
    }
}

// ---------------- 1) h = X @ W via v_wmma_f32_16x16x4_f32 (full f32) ----------
// x: [B, CIN, L] f32 ; W: [CIN, 192] f32 ; h_ws: [N, 192] f32
// Block = 12 waves; block handles 16 consecutive nodes; wave w owns cols [16w,16w+16).
__global__ __launch_bounds__(384) void gat_gemm_wmma(
    const float* __restrict__ x, const float* __restrict__ W,
    float* __restrict__ h_ws)
{
    const int tile      = blockIdx.x;        // 0..2047
    const int node_base = tile << 4;
    const int b         = node_base >> 12;   // node_base / L
    const int l_base    = node_base & (LL - 1);
    const int wave      = threadIdx.x >> 5;  // 0..11
    const int lane      = threadIdx.x & 31;
    const int m         = lane & 15;
    const int g         = lane >> 4;         // half-group
    const int col_base  = wave << 4;

    const float* xb = x + (size_t)b * (CIN * LL) + l_base + m;
    const float* wb = W + col_base + m;

    v8f c = {};
#pragma unroll
    for (int k0 = 0; k0 < CIN; k0 += 4) {
        // A 16x4 f32: lane (m+16g), VGPR j -> A[m][K = 2g + j + k0]
        v2f a, bm;
        const int ka = k0 + (g << 1);
        a.x = xb[(size_t)(ka + 0) * LL];
        a.y = xb[(size_t)(ka + 1) * LL];
        // B 4x16 f32: lane (n+16g), VGPR j -> B[K = 2g + j + k0][n], n == m
        bm.x = wb[(size_t)(ka + 0) * HC];
        bm.y = wb[(size_t)(ka + 1) * HC];
        c = __builtin_amdgcn_wmma_f32_16x16x4_f32(
                /*neg_a=*/false, a, /*neg_b=*/false, bm,
                /*c_mod=*/(short)0, c, /*reuse_a=*/false, /*reuse_b=*/false);
    }

    // D lane (n + 16g), VGPR r -> row r + 8g, col n
    float* out = h_ws + (size_t)(node_base + (g << 3)) * HC + col_base + m;
#pragma unroll
    for (int r = 0; r < 8; ++r)
        out[(size_t)r * HC] = c[r];
}

// ---------------- 2) attention logits a_src/a_dst [N,3] -----------------------
__global__ void gat_att(const float* __restrict__ h_ws,
                        const float* __restrict__ att_src,
                        const float* __restrict__ att_dst,
                        float* __restrict__ a_src, float* __restrict__ a_dst)
{
    int i = blockIdx.x * blockDim.x + threadIdx.x;
    if (i >= NN * HEADS) return;
    int n = i / HEADS, hh = i - n * HEADS;
    const float* hp = h_ws + (size_t)n * HC + hh * COUT;
    const float* as = att_src + hh * COUT;
    const float* ad = att_dst + hh * COUT;
    float ss = 0.f, sd = 0.f;
#pragma unroll 8
    for (int c = 0; c < COUT; ++c) { float v = hp[c]; ss += v * as[c]; sd += v * ad[c]; }
    a_src[i] = ss;
    a_dst[i] = sd;
}

// ---------------- 3) init: emax = self-loop logit; zero denom/acc -------------
__global__ void gat_init(const float* __restrict__ a_src, const float* __restrict__ a_dst,
                         float* __restrict__ emax, float* __restrict__ denom,
                         float* __restrict__ acc)
{
    int i = blockIdx.x * blockDim.x + threadIdx.x;
    const size_t total = (size_t)NN * HC;
    for (size_t j = i; j < total; j += (size_t)gridDim.x * blockDim.x) acc[j] = 0.f;
    if (i < NN * HEADS) {
        float e = a_src[i] + a_dst[i];
        e = e > 0.f ? e : NEG * e;
        emax[i] = e;       // self loop guarantees every segment non-empty
        denom[i] = 0.f;
    }
}

// ---------------- 4) segment max over real edges ------------------------------
__global__ void gat_edge_max(const int* __restrict__ src, const int* __restrict__ dst,
                             const float* __restrict__ a_src, const float* __restrict__ a_dst,
                             float* __restrict__ emax)
{
    int e = blockIdx.x * blockDim.x + threadIdx.x;
    if (e >= EE) return;
    int s = src[e], d = dst[e];
#pragma unroll
    for (int hh = 0; hh < HEADS; ++hh) {
        float v = a_src[s * HEADS + hh] + a_dst[d * HEADS + hh];
        v = v > 0.f ? v : NEG * v;
        atomicMaxFloat(&emax[d * HEADS + hh], v);
    }
}

// ---------------- 5) softmax denominator (edges + self loops) -----------------
__global__ void gat_denom(const int* __restrict__ src, const int* __restrict__ dst,
                          const float* __restrict__ a_src, const float* __restrict__ a_dst,
                          const float* __restrict__ emax, float* __restrict__ denom)
{
    int i = blockIdx.x * blockDim.x + threadIdx.x;
    if (i >= EE + NN) return;
    int s, d;
    if (i < EE) { s = src[i]; d = dst[i]; } else { s = d = i - EE; }
#pragma unroll
    for (int hh = 0; hh < HEADS; ++hh) {
        float v = a_src[s * HEADS + hh] + a_dst[d * HEADS + hh];
        v = v > 0.f ? v : NEG * v;
        atomicAdd(&denom[d * HEADS + hh], __expf(v - emax[d * HEADS + hh]));
    }
}

// ---------------- 6) weighted message scatter: wave per (edge|self) -----------
__global__ void gat_aggregate(const int* __restrict__ src, const int* __restrict__ dst,
                              const float* __restrict__ a_src, const float* __restrict__ a_dst,
                              const float* __restrict__ emax, const float* __restrict__ denom,
                              const float* __restrict__ h_ws, float* __restrict__ acc)
{
    int wid  = (blockIdx.x * blockDim.x + threadIdx.x) >> 5;
    int lane = threadIdx.x & 31;
    if (wid >= EE + NN) return;
    int s, d;
    if (wid < EE) { s = src[wid]; d = dst[wid]; } else { s = d = wid - EE; }
    float alpha[HEADS];
#pragma unroll
    for (int hh = 0; hh < HEADS; ++hh) {
        float v = a_src[s * HEADS + hh] + a_dst[d * HEADS + hh];
        v = v > 0.f ? v : NEG * v;
        alpha[hh] = __expf(v - emax[d * HEADS + hh]) / denom[d * HEADS + hh];
    }
    const float* hs = h_ws + (size_t)s * HC;
    float*       ac = acc  + (size_t)d * HC;
#pragma unroll
    for (int t = 0; t < HC / 32; ++t) {
        int j = lane + (t << 5);
        atomicAdd(&ac[j], alpha[j >> 6] * hs[j]);   // j>>6 is constant per t
    }
}

// ---------------- 7) head mean + bias -----------------------------------------
__global__ void gat_mean_bias(const float* __restrict__ acc, const float* __restrict__ bias,
                              float* __restrict__ outm)
{
    int i = blockIdx.x * blockDim.x + threadIdx.x;   // N*64
    if (i >= NN * COUT) return;
    int n = i >> 6, c = i & 63;
    const float* a = acc + (size_t)n * HC + c;
    outm[i] = (a[0] + a[COUT] + a[2 * COUT]) * (1.0f / 3.0f) + bias[c];
}

// ---------------- 8) per-channel batch stats (64 blocks, LDS tree) ------------
__global__ __launch_bounds__(256) void gat_stats(const float* __restrict__ outm,
                                                 float* __restrict__ stats)
{
    __shared__ float ssum[256], ssq[256];
    const int c = blockIdx.x;
    float s = 0.f, q = 0.f;
    for (int n = threadIdx.x; n < NN; n += 256) {
        float v = outm[(size_t)n * COUT + c];
        s += v; q += v * v;
    }
    ssum[threadIdx.x] = s; ssq[threadIdx.x] = q;
    __syncthreads();
    for (int off = 128; off > 0; off >>= 1) {
        if (threadIdx.x < off) {
            ssum[threadIdx.x] += ssum[threadIdx.x + off];
            ssq[threadIdx.x]  += ssq[threadIdx.x + off];
        }
        __syncthreads();
    }
    if (threadIdx.x == 0) { stats[c] = ssum[0]; stats[COUT + c] = ssq[0]; }
}

// ---------------- 9) BN + ReLU + transpose write to [B, COUT, L] --------------
__global__ void gat_final(const float* __restrict__ outm, const float* __restrict__ stats,
                          const float* __restrict__ gamma, const float* __restrict__ beta,
                          float* __restrict__ out)
{
    int i = blockIdx.x * blockDim.x + threadIdx.x;   // B*COUT*L
    if (i >= NB * COUT * LL) return;
    int l = i & (LL - 1);
    int c = (i >> 12) & 63;
    int b = i >> 18;
    float mu   = stats[c] * (1.0f / NN);
    float var  = stats[COUT + c] * (1.0f / NN) - mu * mu;
    float rstd = rsqrtf(var + 1e-5f);
    float v = outm[(size_t)(b * LL + l) * COUT + c];
    v = (v - mu) * rstd * gamma[c] + beta[c];
    out[i] = v > 0.f ? v : 0.f;
}

// ------------------------------------------------------------------------------
extern "C" void kernel_launch(void* const* d_in, const int* in_sizes, int n_in,
                              void* d_out, int out_size, void* d_ws, size_t ws_size,
                              hipStream_t stream) {
    const float* x       = (const float*)d_in[0];
    // d_in[1] time_emb: dead in reference forward
    const int*   ei      = (const int*)d_in[2];
    const float* W_lin   = (const float*)d_in[3];
    const float* att_src = (const float*)d_in[4];
    const float* att_dst = (const float*)d_in[5];
    const float* bias    = (const float*)d_in[6];
    // d_in[7] tp_W, d_in[8] tp_b: dead
    const float* gamma   = (const float*)d_in[9];
    const float* beta    = (const float*)d_in[10];
    float* out = (float*)d_out;

    const int* src = ei;        // edge_index[0]
    const int* dst = ei + EE;   // edge_index[1]

    // workspace layout (floats)
    float* ws    = (float*)d_ws;
    float* h_ws  = ws;                               // N*192
    float* a_src = h_ws  + (size_t)NN * HC;          // N*3
    float* a_dst = a_src + (size_t)NN * HEADS;       // N*3
    float* emax  = a_dst + (size_t)NN * HEADS;       // N*3
    float* denom = emax  + (size_t)NN * HEADS;       // N*3
    float* acc   = denom + (size_t)NN * HEADS;       // N*192
    float* outm  = acc   + (size_t)NN * HC;          // N*64
    float* stats = outm  + (size_t)NN * COUT;        // 128

    // 1) WMMA GEMM: 2048 node-tiles, 12 waves/block, 16 chained f32 WMMAs
    gat_gemm_wmma<<<NN / 16, 384, 0, stream>>>(x, W_lin, h_ws);
    // 2) attention logits
    gat_att<<<(NN * HEADS + 255) / 256, 256, 0, stream>>>(h_ws, att_src, att_dst, a_src, a_dst);
    // 3) init emax with self loops, zero denom/acc
    gat_init<<<(NN * HEADS + 255) / 256, 256, 0, stream>>>(a_src, a_dst, emax, denom, acc);
    // 4) segment max over edges
    gat_edge_max<<<(EE + 255) / 256, 256, 0, stream>>>(src, dst, a_src, a_dst, emax);
    // 5) softmax denominator
    gat_denom<<<(EE + NN + 255) / 256, 256, 0, stream>>>(src, dst, a_src, a_dst, emax, denom);
    // 6) scatter messages: one wave per edge/self-loop
    {
        long long items = (long long)(EE + NN);
        long long threads = items * 32;
        int blocks = (int)((threads + 255) / 256);
        gat_aggregate<<<blocks, 256, 0, stream>>>(src, dst, a_src, a_dst, emax, denom, h_ws, acc);
    }
    // 7) head mean + bias
    gat_mean_bias<<<(NN * COUT + 255) / 256, 256, 0, stream>>>(acc, bias, outm);
    // 8) batch stats
    gat_stats<<<COUT, 256, 0, stream>>>(outm, stats);
    // 9) BN + ReLU + layout transpose
    gat_final<<<(NB * COUT * LL + 255) / 256, 256, 0, stream>>>(outm, stats, gamma, beta, out);
}